// Generator_85779086836259
// MI455X (gfx1250) — compile-verified
//
#include <hip/hip_runtime.h>
#include <hip/hip_bf16.h>
#include <math.h>

#define BB 4
#define N0 1024
#define NN 2048
#define CC 128
#define KK 20
#define EPSF 1e-5f

typedef float v2f __attribute__((ext_vector_type(2)));
typedef float v8f __attribute__((ext_vector_type(8)));

__device__ __forceinline__ float lrelu_f(float x) { return x >= 0.0f ? x : 0.2f * x; }

// Packed activation layout ("pair-interleaved"): v2f element [p*NN + n] holds
// {x[2p, n], x[2p+1, n]}. One b64 load == one WMMA fragment half per lane.

// ---------------------------------------------------------------------------
// kNN over pc (3 channels, 1024 pts). Keep 21 nearest (ascending d, stable on
// ties to match jax.lax.top_k), drop nearest (self).
// ---------------------------------------------------------------------------
__global__ void knn_pc_kernel(const float* __restrict__ pc, int* __restrict__ idx_out) {
    int t = blockIdx.x * blockDim.x + threadIdx.x;
    if (t >= BB * N0) return;
    int b = t / N0, i = t % N0;
    const float* p = pc + (size_t)b * 3 * N0;
    float xi = p[i], yi = p[N0 + i], zi = p[2 * N0 + i];
    float sqi = xi * xi + yi * yi + zi * zi;
    float dist[KK + 1];
    int   ind[KK + 1];
    for (int q = 0; q <= KK; q++) { dist[q] = 3.4e38f; ind[q] = 0; }
    for (int j = 0; j < N0; j++) {
        float xj = p[j], yj = p[N0 + j], zj = p[2 * N0 + j];
        float d = sqi - 2.0f * (xi * xj + yi * yj + zi * zj) + (xj * xj + yj * yj + zj * zj);
        if (d < dist[KK]) {
            int q = KK;
            while (q > 0 && d < dist[q - 1]) { dist[q] = dist[q - 1]; ind[q] = ind[q - 1]; q--; }
            dist[q] = d; ind[q] = j;
        }
    }
    int* o = idx_out + ((size_t)b * N0 + i) * KK;
    for (int q = 0; q < KK; q++) o[q] = ind[q + 1];   // drop self
}

// ---------------------------------------------------------------------------
// upsample_knn: out[:, :, 0:N0] = x ; out[:, :, N0:2N0] = mean of K neighbors.
// packed=1 writes pair-interleaved layout (Cc must be even).
// ---------------------------------------------------------------------------
__global__ void upsample_kernel(const float* __restrict__ xin, const int* __restrict__ idx,
                                float* __restrict__ xout, int Cc, int packed) {
    int t = blockIdx.x * blockDim.x + threadIdx.x;
    if (t >= BB * Cc * N0) return;
    int n = t % N0;
    int c = (t / N0) % Cc;
    int b = t / (N0 * Cc);
    const float* row = xin + ((size_t)b * Cc + c) * N0;
    const int* id = idx + ((size_t)b * N0 + n) * KK;
    float v = row[n];
    float acc = 0.0f;
    for (int q = 0; q < KK; q++) acc += row[id[q]];
    acc *= (1.0f / (float)KK);
    float* ob = xout + (size_t)b * Cc * NN;
    if (packed) {
        size_t base = ((size_t)(c >> 1) * NN) * 2 + (c & 1);
        ob[base + 2 * (size_t)n] = v;
        ob[base + 2 * (size_t)(N0 + n)] = acc;
    } else {
        ob[(size_t)c * NN + n] = v;
        ob[(size_t)c * NN + N0 + n] = acc;
    }
}

// ---------------------------------------------------------------------------
// Squared norms per point from the packed layout.
// ---------------------------------------------------------------------------
__global__ void sqnorm_pk_kernel(const v2f* __restrict__ xp, float* __restrict__ sq, int Cpairs) {
    int t = blockIdx.x * blockDim.x + threadIdx.x;
    if (t >= BB * NN) return;
    int b = t / NN, n = t % NN;
    const v2f* xb = xp + (size_t)b * Cpairs * NN;
    float s = 0.0f;
    for (int p = 0; p < Cpairs; p++) {
        v2f v = xb[(size_t)p * NN + n];
        s += v[0] * v[0] + v[1] * v[1];
    }
    sq[t] = s;
}

// ---------------------------------------------------------------------------
// Distance matrix via V_WMMA_F32_16X16X4_F32.  16(rows) x 64(cols) per wave:
// A fragment loaded once (b64) feeds 4 WMMAs -> 1.25 b64 loads per WMMA.
// D[m,n] = sq[m] + sq[n] - 2 <x_m, x_n>. Single batch; X (1MB) is L2-resident.
// ---------------------------------------------------------------------------
__global__ void gram_wmma_kernel(const v2f* __restrict__ xp, const float* __restrict__ sqb,
                                 float* __restrict__ D) {
    int wave = blockIdx.x * (blockDim.x >> 5) + (threadIdx.x >> 5);  // 0..4095
    int lane = threadIdx.x & 31;
    int ti = wave >> 5;        // 0..127 row tile
    int tj4 = wave & 31;       // 0..31  group of 4 col tiles
    int lr = lane & 15;
    int hi = lane >> 4;        // selects K-pair {kk,kk+1} vs {kk+2,kk+3}
    int row = ti * 16 + lr;
    int ncol[4];
#pragma unroll
    for (int t = 0; t < 4; t++) ncol[t] = tj4 * 64 + t * 16 + lr;
    v8f acc[4];
#pragma unroll
    for (int t = 0; t < 4; t++) acc[t] = (v8f){0.f,0.f,0.f,0.f,0.f,0.f,0.f,0.f};
    for (int kk = 0; kk < CC; kk += 4) {
        int pi = (kk >> 1) + hi;
        v2f a = xp[(size_t)pi * NN + row];
#pragma unroll
        for (int t = 0; t < 4; t++) {
            v2f bf = xp[(size_t)pi * NN + ncol[t]];
            acc[t] = __builtin_amdgcn_wmma_f32_16x16x4_f32(false, a, false, bf,
                                                           (short)0, acc[t], false, false);
        }
    }
#pragma unroll
    for (int t = 0; t < 4; t++) {
#pragma unroll
        for (int r = 0; r < 8; r++) {
            int m = ti * 16 + r + hi * 8;   // C/D layout: VGPR r = row r (lanes<16) / r+8
            int n = ncol[t];
            D[(size_t)m * NN + n] = sqb[m] + sqb[n] - 2.0f * acc[t][r];
        }
    }
}

// ---------------------------------------------------------------------------
// Row-wise top-K smallest (stable on ties); self included (matches knn_idx(x,k))
// ---------------------------------------------------------------------------
__global__ void topk_kernel(const float* __restrict__ D, int* __restrict__ idx_out) {
    int n = blockIdx.x * blockDim.x + threadIdx.x;
    if (n >= NN) return;
    const float* row = D + (size_t)n * NN;
    float dist[KK];
    int   ind[KK];
    for (int q = 0; q < KK; q++) { dist[q] = 3.4e38f; ind[q] = 0; }
    for (int j = 0; j < NN; j++) {
        float d = row[j];
        if (d < dist[KK - 1]) {
            int q = KK - 1;
            while (q > 0 && d < dist[q - 1]) { dist[q] = dist[q - 1]; ind[q] = ind[q - 1]; q--; }
            dist[q] = d; ind[q] = j;
        }
    }
    int* o = idx_out + (size_t)n * KK;
    for (int q = 0; q < KK; q++) o[q] = ind[q];
}

// ---------------------------------------------------------------------------
// Fold edgeconv weights into WMMA-ready pair-interleaved layouts:
//   Afold[p][o]={A[2p][o],A[2p+1][o]}, A[c][o]=sum_j(W[o,c,j]-W[o,C+c,j])
//   Wdt[j][p][o]={W[o,C+2p,j], W[o,C+2p+1,j]}
//   scale[o]=g/sqrt(v+eps); shift[o]=(bias-m)*scale+bt   (BN folded)
// ---------------------------------------------------------------------------
__global__ void fold_w_kernel(const float* __restrict__ W, const float* __restrict__ bias,
                              const float* __restrict__ g, const float* __restrict__ btb,
                              const float* __restrict__ mm, const float* __restrict__ vv,
                              float* __restrict__ Afold, float* __restrict__ Wdt,
                              float* __restrict__ scale, float* __restrict__ shift) {
    int t = blockIdx.x * blockDim.x + threadIdx.x;
    if (t < KK * CC * CC) {
        int o = t % CC;
        int c = (t / CC) % CC;
        int j = t / (CC * CC);
        float val = W[((size_t)o * 2 * CC + CC + c) * KK + j];
        Wdt[(((size_t)j * (CC / 2) + (c >> 1)) * CC + o) * 2 + (c & 1)] = val;
    }
    if (t < CC * CC) {
        int o = t % CC;
        int c = t / CC;
        float s = 0.0f;
        for (int j = 0; j < KK; j++)
            s += W[((size_t)o * 2 * CC + c) * KK + j] - W[((size_t)o * 2 * CC + CC + c) * KK + j];
        Afold[(((size_t)(c >> 1)) * CC + o) * 2 + (c & 1)] = s;
    }
    if (t < CC) {
        float sc = g[t] / sqrtf(vv[t] + EPSF);
        scale[t] = sc;
        shift[t] = (bias[t] - mm[t]) * sc + btb[t];
    }
}

// ---------------------------------------------------------------------------
// Fused edgeconv: central GEMM + gather-GEMM over K neighbors, BN + lrelu.
// 16(o) x 64(n) per wave, shared A fragment, b64 gathers, packed b64 stores.
// ---------------------------------------------------------------------------
__global__ void edgeconv_wmma_kernel(const v2f* __restrict__ xp, const int* __restrict__ idx,
                                     const v2f* __restrict__ Afold, const v2f* __restrict__ Wdt,
                                     const float* __restrict__ scale, const float* __restrict__ shift,
                                     v2f* __restrict__ outp) {
    int b = blockIdx.y;
    const v2f* xb = xp + (size_t)b * (CC / 2) * NN;
    const int* idxb = idx + (size_t)b * NN * KK;
    v2f* outb = outp + (size_t)b * (CC / 2) * NN;
    int wave = blockIdx.x * (blockDim.x >> 5) + (threadIdx.x >> 5);  // 0..255
    int lane = threadIdx.x & 31;
    int t_o = wave >> 5;     // 0..7
    int t_n4 = wave & 31;    // 0..31
    int lr = lane & 15, hi = lane >> 4;
    int o16 = t_o * 16 + lr;
    int ncol[4];
#pragma unroll
    for (int t = 0; t < 4; t++) ncol[t] = t_n4 * 64 + t * 16 + lr;
    v8f acc[4];
#pragma unroll
    for (int t = 0; t < 4; t++) acc[t] = (v8f){0.f,0.f,0.f,0.f,0.f,0.f,0.f,0.f};
    // central term: A(128x128) x X(128x2048)
    for (int kk = 0; kk < CC; kk += 4) {
        int pi = (kk >> 1) + hi;
        v2f a = Afold[(size_t)pi * CC + o16];
#pragma unroll
        for (int t = 0; t < 4; t++) {
            v2f bf = xb[(size_t)pi * NN + ncol[t]];
            acc[t] = __builtin_amdgcn_wmma_f32_16x16x4_f32(false, a, false, bf,
                                                           (short)0, acc[t], false, false);
        }
    }
    // neighbor terms: gather fused into B-fragment b64 loads
    for (int j = 0; j < KK; j++) {
        int gi[4];
#pragma unroll
        for (int t = 0; t < 4; t++) gi[t] = idxb[(size_t)ncol[t] * KK + j];
        const v2f* Wj = Wdt + (size_t)j * (CC / 2) * CC;
        for (int kk = 0; kk < CC; kk += 4) {
            int pi = (kk >> 1) + hi;
            v2f a = Wj[(size_t)pi * CC + o16];
#pragma unroll
            for (int t = 0; t < 4; t++) {
                v2f bf = xb[(size_t)pi * NN + gi[t]];
                acc[t] = __builtin_amdgcn_wmma_f32_16x16x4_f32(false, a, false, bf,
                                                               (short)0, acc[t], false, false);
            }
        }
    }
    int base_o = t_o * 16 + hi * 8;     // even, rows r and r+1 are adjacent channels
#pragma unroll
    for (int t = 0; t < 4; t++) {
#pragma unroll
        for (int r = 0; r < 8; r += 2) {
            int o = base_o + r;
            v2f v;
            v[0] = lrelu_f(acc[t][r]     * scale[o]     + shift[o]);
            v[1] = lrelu_f(acc[t][r + 1] * scale[o + 1] + shift[o + 1]);
            outb[(size_t)(o >> 1) * NN + ncol[t]] = v;
        }
    }
}

// ---------------------------------------------------------------------------
// AdaIN instance norm: gamma/beta = style @ Wa^T + ba ; normalize over N.
// One block per (channel, batch). Optional lrelu on input; packed/scalar IO.
// ---------------------------------------------------------------------------
__global__ void instnorm_kernel(const float* __restrict__ xin, float* __restrict__ xout,
                                const float* __restrict__ style, int srow,
                                const float* __restrict__ Wa, const float* __restrict__ ba,
                                int apply_lrelu, int in_packed, int out_packed) {
    int c = blockIdx.x;
    int b = blockIdx.y;
    int tid = threadIdx.x;                 // 256 threads
    __shared__ float red[256];
    __shared__ float red2[256];
    __shared__ float s_gamma, s_beta, s_mean, s_inv;
    const float* s = style + ((size_t)b * 3 + srow) * 128;
    float pg = 0.0f, pb = 0.0f;
    if (tid < 128) {
        float sv = s[tid];
        pg = Wa[(size_t)c * 128 + tid] * sv;
        pb = Wa[((size_t)CC + c) * 128 + tid] * sv;
    }
    red[tid] = pg; red2[tid] = pb;
    __syncthreads();
    for (int st = 128; st > 0; st >>= 1) {
        if (tid < st) { red[tid] += red[tid + st]; red2[tid] += red2[tid + st]; }
        __syncthreads();
    }
    if (tid == 0) { s_gamma = red[0] + ba[c]; s_beta = red2[0] + ba[CC + c]; }

    const float* ib = xin + (size_t)b * CC * NN;
    const float* irow = in_packed ? (ib + (size_t)(c >> 1) * 2 * NN + (c & 1))
                                  : (ib + (size_t)c * NN);
    int istep = in_packed ? 2 : 1;
    float sum = 0.0f, ss = 0.0f;
    for (int n = tid; n < NN; n += 256) {
        float v = irow[(size_t)n * istep];
        if (apply_lrelu) v = lrelu_f(v);
        sum += v; ss += v * v;
    }
    __syncthreads();
    red[tid] = sum; red2[tid] = ss;
    __syncthreads();
    for (int st = 128; st > 0; st >>= 1) {
        if (tid < st) { red[tid] += red[tid + st]; red2[tid] += red2[tid + st]; }
        __syncthreads();
    }
    if (tid == 0) {
        float mu = red[0] / (float)NN;
        float var = red2[0] / (float)NN - mu * mu;
        s_mean = mu;
        s_inv = 1.0f / sqrtf(var + EPSF);
    }
    __syncthreads();
    float gamma = s_gamma, beta = s_beta, mu = s_mean, inv = s_inv;
    float* ob = xout + (size_t)b * CC * NN;
    float* orow = out_packed ? (ob + (size_t)(c >> 1) * 2 * NN + (c & 1))
                             : (ob + (size_t)c * NN);
    int ostep = out_packed ? 2 : 1;
    for (int n = tid; n < NN; n += 256) {
        float v = irow[(size_t)n * istep];
        if (apply_lrelu) v = lrelu_f(v);
        orow[(size_t)n * ostep] = gamma * (v - mu) * inv + beta;
    }
}

// ---------------------------------------------------------------------------
// Generic WMMA GEMM + bias + lrelu with packed IO (16 x 64 slab per wave)
// ---------------------------------------------------------------------------
__global__ void gemm_lrelu_wmma_kernel(const v2f* __restrict__ WTp, const float* __restrict__ bias,
                                       const v2f* __restrict__ xp, v2f* __restrict__ outp,
                                       int Cin, int O) {
    int b = blockIdx.y;
    const v2f* xb = xp + (size_t)b * (Cin / 2) * NN;
    v2f* outb = outp + (size_t)b * (O / 2) * NN;
    int wave = blockIdx.x * (blockDim.x >> 5) + (threadIdx.x >> 5);
    int lane = threadIdx.x & 31;
    int t_o = wave >> 5;
    int t_n4 = wave & 31;
    int lr = lane & 15, hi = lane >> 4;
    int o16 = t_o * 16 + lr;
    int ncol[4];
#pragma unroll
    for (int t = 0; t < 4; t++) ncol[t] = t_n4 * 64 + t * 16 + lr;
    v8f acc[4];
#pragma unroll
    for (int t = 0; t < 4; t++) acc[t] = (v8f){0.f,0.f,0.f,0.f,0.f,0.f,0.f,0.f};
    for (int kk = 0; kk < Cin; kk += 4) {
        int pi = (kk >> 1) + hi;
        v2f a = WTp[(size_t)pi * O + o16];
#pragma unroll
        for (int t = 0; t < 4; t++) {
            v2f bf = xb[(size_t)pi * NN + ncol[t]];
            acc[t] = __builtin_amdgcn_wmma_f32_16x16x4_f32(false, a, false, bf,
                                                           (short)0, acc[t], false, false);
        }
    }
    int base_o = t_o * 16 + hi * 8;
#pragma unroll
    for (int t = 0; t < 4; t++) {
#pragma unroll
        for (int r = 0; r < 8; r += 2) {
            int o = base_o + r;
            v2f v;
            v[0] = lrelu_f(acc[t][r]     + bias[o]);
            v[1] = lrelu_f(acc[t][r + 1] + bias[o + 1]);
            outb[(size_t)(o >> 1) * NN + ncol[t]] = v;
        }
    }
}

// Transpose + pair-interleave a dense weight: WTp[p][o] = {W[o][2p], W[o][2p+1]}
__global__ void transpose_pk_kernel(const float* __restrict__ Win, float* __restrict__ WTp,
                                    int O, int Cin) {
    int t = blockIdx.x * blockDim.x + threadIdx.x;
    if (t >= O * Cin) return;
    int o = t / Cin, c = t % Cin;
    WTp[(((size_t)(c >> 1)) * O + o) * 2 + (c & 1)] = Win[(size_t)o * Cin + c];
}

// ---------------------------------------------------------------------------
// Final tiny layer: y3 = lrelu(Wf3 @ m2 + bf3); pc_out = pc2 + y3
// ---------------------------------------------------------------------------
__global__ void final_kernel(const v2f* __restrict__ m2p, const float* __restrict__ Wf3,
                             const float* __restrict__ bf3, const float* __restrict__ pc2,
                             float* __restrict__ pc_out) {
    int t = blockIdx.x * blockDim.x + threadIdx.x;
    if (t >= BB * 3 * NN) return;
    int n = t % NN;
    int o = (t / NN) % 3;
    int b = t / (3 * NN);
    const v2f* xb = m2p + (size_t)b * 32 * NN;
    float acc = bf3[o];
    for (int p = 0; p < 32; p++) {
        v2f v = xb[(size_t)p * NN + n];
        acc += Wf3[o * 64 + 2 * p] * v[0] + Wf3[o * 64 + 2 * p + 1] * v[1];
    }
    acc = lrelu_f(acc);
    pc_out[((size_t)b * 3 + o) * NN + n] = pc2[((size_t)b * 3 + o) * NN + n] + acc;
}

// ---------------------------------------------------------------------------
extern "C" void kernel_launch(void* const* d_in, const int* in_sizes, int n_in,
                              void* d_out, int out_size, void* d_ws, size_t ws_size,
                              hipStream_t stream) {
    const float* content = (const float*)d_in[0];
    const float* pc      = (const float*)d_in[1];
    const float* style   = (const float*)d_in[2];
    const float* W1      = (const float*)d_in[3];
    const float* b1      = (const float*)d_in[4];
    const float* bn1_g   = (const float*)d_in[5];
    const float* bn1_b   = (const float*)d_in[6];
    const float* bn1_m   = (const float*)d_in[7];
    const float* bn1_v   = (const float*)d_in[8];
    const float* Wa1     = (const float*)d_in[9];
    const float* ba1     = (const float*)d_in[10];
    const float* W2      = (const float*)d_in[11];
    const float* b2      = (const float*)d_in[12];
    const float* bn2_g   = (const float*)d_in[13];
    const float* bn2_b   = (const float*)d_in[14];
    const float* bn2_m   = (const float*)d_in[15];
    const float* bn2_v   = (const float*)d_in[16];
    const float* Wa2     = (const float*)d_in[17];
    const float* ba2     = (const float*)d_in[18];
    const float* Wt      = (const float*)d_in[19];
    const float* bt      = (const float*)d_in[20];
    const float* Wf1     = (const float*)d_in[21];
    const float* bf1     = (const float*)d_in[22];
    const float* Wf2     = (const float*)d_in[23];
    const float* bf2     = (const float*)d_in[24];
    const float* Wf3     = (const float*)d_in[25];
    const float* bf3     = (const float*)d_in[26];

    float* x_out  = (float*)d_out;                       // (B, 128, 2048) scalar layout
    float* pc_out = x_out + (size_t)BB * CC * NN;        // (B, 3, 2048)

    char* w = (char*)d_ws;
    auto alloc = [&](size_t bytes) -> char* {
        char* p = w;
        w += (bytes + 255) & ~(size_t)255;
        return p;
    };
    int*   idx_pc = (int*)alloc((size_t)BB * N0 * KK * 4);
    int*   idx_x  = (int*)alloc((size_t)BB * NN * KK * 4);
    float* x_up_p = (float*)alloc((size_t)BB * CC * NN * 4);     // packed
    float* pc2    = (float*)alloc((size_t)BB * 3 * NN * 4);      // scalar
    float* sqn    = (float*)alloc((size_t)BB * NN * 4);
    float* Dbuf   = (float*)alloc((size_t)NN * NN * 4);          // per-batch, reused
    float* Afold1 = (float*)alloc((size_t)CC * CC * 4);          // packed pairs
    float* Wdt1   = (float*)alloc((size_t)KK * CC * CC * 4);     // packed pairs
    float* sc1    = (float*)alloc((size_t)CC * 4);
    float* sh1    = (float*)alloc((size_t)CC * 4);
    float* Afold2 = (float*)alloc((size_t)CC * CC * 4);
    float* Wdt2   = (float*)alloc((size_t)KK * CC * CC * 4);
    float* sc2    = (float*)alloc((size_t)CC * 4);
    float* sh2    = (float*)alloc((size_t)CC * 4);
    float* e1p    = (float*)alloc((size_t)BB * CC * NN * 4);     // packed
    float* h1p    = (float*)alloc((size_t)BB * CC * NN * 4);     // packed
    float* e2p    = (float*)alloc((size_t)BB * CC * NN * 4);     // packed
    float* ybufp  = (float*)alloc((size_t)BB * CC * NN * 4);     // packed
    float* mlp1p  = (float*)alloc((size_t)BB * CC * NN * 4);     // packed
    float* mlp2p  = (float*)alloc((size_t)BB * 64 * NN * 4);     // packed
    float* Wf1Tp  = (float*)alloc((size_t)128 * 128 * 4);
    float* Wf2Tp  = (float*)alloc((size_t)128 * 64 * 4);
    (void)ws_size; (void)in_sizes; (void)n_in; (void)out_size;

    // 1) kNN on pc, upsample content (packed) and pc (scalar)
    knn_pc_kernel<<<(BB * N0 + 255) / 256, 256, 0, stream>>>(pc, idx_pc);
    upsample_kernel<<<(BB * CC * N0 + 255) / 256, 256, 0, stream>>>(content, idx_pc, x_up_p, CC, 1);
    upsample_kernel<<<(BB * 3 * N0 + 255) / 256, 256, 0, stream>>>(pc, idx_pc, pc2, 3, 0);

    // 2) kNN on x_up (WMMA Gram + top-k), per batch reusing D buffer
    sqnorm_pk_kernel<<<(BB * NN + 255) / 256, 256, 0, stream>>>((const v2f*)x_up_p, sqn, CC / 2);
    for (int b = 0; b < BB; b++) {
        gram_wmma_kernel<<<512, 256, 0, stream>>>(
            (const v2f*)x_up_p + (size_t)b * (CC / 2) * NN, sqn + b * NN, Dbuf);
        topk_kernel<<<(NN + 255) / 256, 256, 0, stream>>>(Dbuf, idx_x + (size_t)b * NN * KK);
    }

    // 3) edgeconv1 (fused gather-GEMM, BN, lrelu) then AdaIN (with input lrelu)
    fold_w_kernel<<<(KK * CC * CC + 255) / 256, 256, 0, stream>>>(
        W1, b1, bn1_g, bn1_b, bn1_m, bn1_v, Afold1, Wdt1, sc1, sh1);
    edgeconv_wmma_kernel<<<dim3(32, BB), 256, 0, stream>>>(
        (const v2f*)x_up_p, idx_x, (const v2f*)Afold1, (const v2f*)Wdt1, sc1, sh1, (v2f*)e1p);
    instnorm_kernel<<<dim3(CC, BB), 256, 0, stream>>>(e1p, h1p, style, 0, Wa1, ba1, 1, 1, 1);

    // 4) kNN on h1, edgeconv2, AdaIN -> x output (scalar layout in d_out)
    sqnorm_pk_kernel<<<(BB * NN + 255) / 256, 256, 0, stream>>>((const v2f*)h1p, sqn, CC / 2);
    for (int b = 0; b < BB; b++) {
        gram_wmma_kernel<<<512, 256, 0, stream>>>(
            (const v2f*)h1p + (size_t)b * (CC / 2) * NN, sqn + b * NN, Dbuf);
        topk_kernel<<<(NN + 255) / 256, 256, 0, stream>>>(Dbuf, idx_x + (size_t)b * NN * KK);
    }
    fold_w_kernel<<<(KK * CC * CC + 255) / 256, 256, 0, stream>>>(
        W2, b2, bn2_g, bn2_b, bn2_m, bn2_v, Afold2, Wdt2, sc2, sh2);
    edgeconv_wmma_kernel<<<dim3(32, BB), 256, 0, stream>>>(
        (const v2f*)h1p, idx_x, (const v2f*)Afold2, (const v2f*)Wdt2, sc2, sh2, (v2f*)e2p);
    instnorm_kernel<<<dim3(CC, BB), 256, 0, stream>>>(e2p, x_out, style, 1, Wa2, ba2, 1, 1, 0);

    // 5) third AdaIN (no input lrelu) -> packed y, MLP tail, pc_out = pc2 + y3
    instnorm_kernel<<<dim3(CC, BB), 256, 0, stream>>>(x_out, ybufp, style, 2, Wt, bt, 0, 0, 1);
    transpose_pk_kernel<<<(128 * 128 + 255) / 256, 256, 0, stream>>>(Wf1, Wf1Tp, 128, 128);
    transpose_pk_kernel<<<(64 * 128 + 255) / 256, 256, 0, stream>>>(Wf2, Wf2Tp, 64, 128);
    gemm_lrelu_wmma_kernel<<<dim3(32, BB), 256, 0, stream>>>(
        (const v2f*)Wf1Tp, bf1, (const v2f*)ybufp, (v2f*)mlp1p, 128, 128);
    gemm_lrelu_wmma_kernel<<<dim3(16, BB), 256, 0, stream>>>(
        (const v2f*)Wf2Tp, bf2, (const v2f*)mlp1p, (v2f*)mlp2p, 128, 64);
    final_kernel<<<(BB * 3 * NN + 255) / 256, 256, 0, stream>>>(
        (const v2f*)mlp2p, Wf3, bf3, pc2, pc_out);
}